// HyperDistance_61701500174341
// MI455X (gfx1250) — compile-verified
//
#include <hip/hip_runtime.h>

// HyperDistance on MI455X (gfx1250, wave32).
//
// Roofline: 1 GiB read-once stream (2 x [2M,64] f32), ~0.4 GFLOP of math.
// HBM-bound: floor ~46us @ 23.3 TB/s. Strategy:
//   - non-temporal loads (stream > 192MB L2, each byte used once)
//   - per-wave 16-row tiles; row-wise Lorentz inner products computed on the
//     matrix pipe via V_WMMA_F32_16X16X4_F32 (diag of (metric*B) @ T^T tile),
//     16 accumulating WMMA steps cover K=0..63
//   - sched_barrier(0) splits the tile into a pure load phase (32 b64 NT
//     loads in flight -> high per-wave MLP) and a pure WMMA phase
//   - arccosh + deterministic reduction (shfl -> LDS -> partials in d_ws ->
//     final 1-block reduce writes mean to d_out)

typedef float v2f __attribute__((ext_vector_type(2)));
typedef float v8f __attribute__((ext_vector_type(8)));

#define DIM 64
#define TILE_ROWS 16

__global__ __launch_bounds__(256) void hyper_dist_partial_kernel(
    const float* __restrict__ batch,   // [n, 64]
    const float* __restrict__ target,  // [n, 64]
    const float* __restrict__ metric,  // [64]
    float* __restrict__ partials,      // [gridDim.x]
    int numTiles)
{
  const int lane = threadIdx.x & 31;
  const int wave = threadIdx.x >> 5;
  const int wavesPerBlock = blockDim.x >> 5;
  const int gwave = blockIdx.x * wavesPerBlock + wave;
  const int nwaves = gridDim.x * wavesPerBlock;

  const int half = lane >> 4;        // 0: lanes 0-15, 1: lanes 16-31
  const int rowInTile = lane & 15;

  // Per-lane metric float2 for each of the 16 K-steps (columns 4j+2*half, +1).
  v2f mv[16];
#pragma unroll
  for (int j = 0; j < 16; ++j)
    mv[j] = *(const v2f*)(metric + 4 * j + 2 * half);

  float acc = 0.0f;

  for (int tile = gwave; tile < numTiles; tile += nwaves) {
    const long off0 = ((long)tile * TILE_ROWS + rowInTile) * DIM + 2 * half;

    // Phase 1: issue ALL tile loads (32 x global_load_b64, NT).
    v2f a[16], b[16];
#pragma unroll
    for (int j = 0; j < 16; ++j) {
      a[j] = __builtin_nontemporal_load((const v2f*)(batch  + off0 + 4 * j));
      b[j] = __builtin_nontemporal_load((const v2f*)(target + off0 + 4 * j));
    }

    // Hard scheduling fence: nothing crosses. Keeps all 32 loads issued
    // up-front (8KiB in flight per wave) instead of being sunk into the
    // WMMA chain by the backend scheduler.
    __builtin_amdgcn_sched_barrier(0);

    // Phase 2: 16 accumulating WMMA steps.
    // WMMA f32 16x16x4 per-lane layout: lane l supplies A[l&15][k0+2*(l>>4)+{0,1}]
    // in (VGPR0,VGPR1); B (=T-tile^T) has the identical per-lane pattern.
    v8f c = {0.f, 0.f, 0.f, 0.f, 0.f, 0.f, 0.f, 0.f};
#pragma unroll
    for (int j = 0; j < 16; ++j) {
      v2f am = a[j] * mv[j];  // fold metric into A
      c = __builtin_amdgcn_wmma_f32_16x16x4_f32(
          false, am, false, b[j], (short)0, c, false, false);
    }

    // Diagonal of the 16x16 f32 accumulator: row M=v in VGPR v (lanes 0-15,
    // N=lane) and row M=v+8 in VGPR v (lanes 16-31, N=lane-16). Diag m==n is
    // at (VGPR v, lane v) and (VGPR v, lane v+24). Other lanes keep 0 ->
    // arccosh(1) = 0, contributing nothing.
    float dv = 0.0f;
#pragma unroll
    for (int v = 0; v < 8; ++v) {
      bool sel = (lane == v) || (lane == v + 24);
      dv = sel ? c[v] : dv;
    }

    float x = fmaxf(-dv, 1.0f);
    acc += logf(x + sqrtf(x * x - 1.0f));   // arccosh(x)
  }

  // Wave reduce (wave32).
#pragma unroll
  for (int off = 16; off > 0; off >>= 1)
    acc += __shfl_xor(acc, off, 32);

  __shared__ float wsum[8];
  if (lane == 0) wsum[wave] = acc;
  __syncthreads();
  if (wave == 0) {
    float s = (lane < wavesPerBlock) ? wsum[lane] : 0.0f;
#pragma unroll
    for (int off = 16; off > 0; off >>= 1)
      s += __shfl_xor(s, off, 32);
    if (lane == 0) partials[blockIdx.x] = s;  // written by EVERY block
  }
}

__global__ __launch_bounds__(256) void hyper_dist_finalize_kernel(
    const float* __restrict__ partials, int np,
    float* __restrict__ out, float invN)
{
  const int lane = threadIdx.x & 31;
  const int wave = threadIdx.x >> 5;

  float s = 0.0f;
  for (int i = threadIdx.x; i < np; i += blockDim.x) s += partials[i];

#pragma unroll
  for (int off = 16; off > 0; off >>= 1)
    s += __shfl_xor(s, off, 32);

  __shared__ float wsum[8];
  if (lane == 0) wsum[wave] = s;
  __syncthreads();
  if (wave == 0) {
    float t = (lane < (int)(blockDim.x >> 5)) ? wsum[lane] : 0.0f;
#pragma unroll
    for (int off = 16; off > 0; off >>= 1)
      t += __shfl_xor(t, off, 32);
    if (lane == 0) out[0] = t * invN;
  }
}

extern "C" void kernel_launch(void* const* d_in, const int* in_sizes, int n_in,
                              void* d_out, int out_size, void* d_ws, size_t ws_size,
                              hipStream_t stream) {
  const float* batch  = (const float*)d_in[0];
  const float* target = (const float*)d_in[1];
  const float* metric = (const float*)d_in[2];
  float* out = (float*)d_out;
  float* partials = (float*)d_ws;

  const long n = (long)in_sizes[0] / DIM;        // 2,097,152
  const int numTiles = (int)(n / TILE_ROWS);     // n is a multiple of 16
  const int blocks = 2048;                       // 16384 waves, 8 tiles/wave

  hyper_dist_partial_kernel<<<blocks, 256, 0, stream>>>(
      batch, target, metric, partials, numTiles);
  hyper_dist_finalize_kernel<<<1, 256, 0, stream>>>(
      partials, blocks, out, 1.0f / (float)n);
}